// Model_9612136808478
// MI455X (gfx1250) — compile-verified
//
#include <hip/hip_runtime.h>
#include <cmath>

typedef __attribute__((ext_vector_type(2))) float v2f;
typedef __attribute__((ext_vector_type(4))) float v4f;
typedef __attribute__((ext_vector_type(8))) float v8f;

__device__ __forceinline__ v8f wmma4(v2f a, v2f b, v8f c) {
  // D = A(16x4) * B(4x16) + C, fp32
  return __builtin_amdgcn_wmma_f32_16x16x4_f32(false, a, false, b, (short)0, c, false, false);
}

__device__ __forceinline__ float rcp_fast(float x) { return __builtin_amdgcn_rcpf(x); }
__device__ __forceinline__ float sigm(float x) { return rcp_fast(1.f + __expf(-x)); }
// tanh(x) = 1 - 2/(exp(2x)+1): saturates correctly at +/-inf, ~5 VALU ops
__device__ __forceinline__ float tanh_fast(float x) { return 1.f - 2.f * rcp_fast(__expf(2.f * x) + 1.f); }

// ---------------- LDS layout (floats) ----------------
// 7 uniform stacks, each W_STK floats (fragment-interleaved, zero padded):
//   wih0@0    : 6 kg * 3 nt * 64 = 1152   (K padded to 24)
//   whh0@1152 : 576                       (K=12; row 10 = bias0, row 11 = 0)
//   wih1@1728 : 576                       (rows 10,11 = 0)
//   whh1@2304 : 576                       (row 10 = bias1, row 11 = 0)
#define W_STK 2880
#define W_TOT 20336 // 7*2880 + 176 (fc1_w@20160[100], fc1_b@20260[10], fc2_w@20270[60], fc2_b@20330[6])
// Per-wave region:
//   gatesA@0 [16][52]=832 ; gatesB@832 ; hA_l0[s]@1664+s*192 (stride 12)
//   concatU@2432 [16][24] ; concatL@2816 ; hB_l0[u]@3200+u*192
//   concatBody@3584 [16][24] ; hBody_l0@3968 [16][12] ; hBody_l1@4160 [16][12]
// Every h-state row carries a constant 1.0 at its pad column 10 (and 22 for concat
// buffers) so the recurrent WMMA adds the bias stored in whh row 10.
#define GSTR 52
#define PW_SZ 4352
#define LDS_FLOATS (W_TOT + 4 * PW_SZ) // 37744
#define LDS_BYTES (LDS_FLOATS * 4)     // 150976 -> 2 blocks/WGP within 320KB

// Fragment-interleaved index for a B operand element at (physical K row k, physical N col n):
// v2f lane fetch = floats [(kg*3+nt)*64 + lane*2 + {0,1}]
__device__ __forceinline__ int fragIdx(int k, int n) {
  return ((k >> 2) * 3 + (n >> 4)) * 64 + (((k & 2) << 3) + (n & 15)) * 2 + (k & 1);
}
// Gate interleave: reference gate index n (i:0-9,f:10-19,g:20-29,o:30-39) -> column 4*j + type
__device__ __forceinline__ int permN(int n) { return 4 * (n % 10) + n / 10; }

// One LSTM cell step for a 16-row batch tile, wave-cooperative.
template <int KGIN, typename AF>
__device__ __forceinline__ void lstm_cell(
    AF afetch,                                  // afetch(kg, v2f& a): A fragment for input matmul
    const float* __restrict__ wih,              // LDS fragment-interleaved
    const float* __restrict__ whh,              // LDS fragment-interleaved (row 10 = bias)
    const float* hprev, int hstride, int hoff,  // recurrent h source (may alias hout; col10 = 1.0)
    float* __restrict__ gates,                  // LDS [16][GSTR] scratch
    float* hout, int hostride, int hooff,       // h destination
    float* c5, int row, int hi, int lane)
{
  v8f acc[3];
#pragma unroll
  for (int nt = 0; nt < 3; ++nt)
    acc[nt] = (v8f){0.f, 0.f, 0.f, 0.f, 0.f, 0.f, 0.f, 0.f}; // lowered to inline SRC2=0
  const v2f* wi = (const v2f*)wih;
#pragma unroll
  for (int kg = 0; kg < KGIN; ++kg) {
    v2f a;
    afetch(kg, a);
#pragma unroll
    for (int nt = 0; nt < 3; ++nt)
      acc[nt] = wmma4(a, wi[(kg * 3 + nt) * 32 + lane], acc[nt]);
  }
  const v2f* wh = (const v2f*)whh;
#pragma unroll
  for (int kg = 0; kg < 3; ++kg) { // recurrent K=12: [h0..h9, 1.0, 0] x [Whh; bias; 0]
    v2f a = *(const v2f*)(hprev + row * hstride + hoff + 4 * kg + 2 * hi);
#pragma unroll
    for (int nt = 0; nt < 3; ++nt)
      acc[nt] = wmma4(a, wh[(kg * 3 + nt) * 32 + lane], acc[nt]);
  }
  // spill D tiles -> gates[M=batch][N=permuted gate]
#pragma unroll
  for (int nt = 0; nt < 3; ++nt)
#pragma unroll
    for (int r = 0; r < 8; ++r)
      gates[(r + 8 * hi) * GSTR + nt * 16 + row] = acc[nt][r];
  // pointwise: lane(row,hi) handles hidden j = hi*5 .. hi*5+4; i,f,g,o contiguous (b128 load)
  const int j0 = hi * 5;
#pragma unroll
  for (int jj = 0; jj < 5; ++jj) {
    v4f g4 = *(const v4f*)(gates + row * GSTR + 4 * (j0 + jj));
    float iv = sigm(g4.x);
    float fv = sigm(g4.y);
    float gv = tanh_fast(g4.z);
    float ov = sigm(g4.w);
    float cv = fv * c5[jj] + iv * gv;
    c5[jj] = cv;
    hout[row * hostride + hooff + j0 + jj] = ov * tanh_fast(cv);
  }
}

__global__ __launch_bounds__(128, 1) void lstm_tree_kernel(
    const float* __restrict__ x,
    const float* __restrict__ wa0_ih, const float* __restrict__ wa1_ih,
    const float* __restrict__ wa_hh, const float* __restrict__ ba_ih,
    const float* __restrict__ ba_hh,
    const float* __restrict__ wb0_ih, const float* __restrict__ wb1_ih,
    const float* __restrict__ wb_hh, const float* __restrict__ bb_ih,
    const float* __restrict__ bb_hh,
    const float* __restrict__ fc1_w, const float* __restrict__ fc1_b,
    const float* __restrict__ fc2_w, const float* __restrict__ fc2_b,
    float* __restrict__ out)
{
  extern __shared__ float lds[];
  const int tid = threadIdx.x;
  const int bd = 128;

  // zero all of LDS: weight pads + per-wave h state for t=0
  for (int i = tid; i < LDS_FLOATS; i += bd) lds[i] = 0.f;
  __syncthreads();

  // ---- stage weights, fragment-interleaved + gate-permuted ----
  for (int i = tid; i < 4 * 40 * 22; i += bd) { // wa0_ih (4,40,22)
    int s = i / 880, r = i % 880, n = r / 22, k = r % 22;
    lds[s * W_STK + fragIdx(k, permN(n))] = wa0_ih[i];
  }
  for (int i = tid; i < 4 * 40 * 10; i += bd) { // wa1_ih (4,40,10)
    int s = i / 400, r = i % 400, n = r / 10, k = r % 10;
    lds[s * W_STK + 1728 + fragIdx(k, permN(n))] = wa1_ih[i];
  }
  for (int i = tid; i < 4 * 2 * 40 * 10; i += bd) { // wa_hh (4,2,40,10)
    int s = i / 800, r = i % 800, l = r / 400, n = (r % 400) / 10, k = r % 10;
    lds[s * W_STK + (l ? 2304 : 1152) + fragIdx(k, permN(n))] = wa_hh[i];
  }
  for (int i = tid; i < 4 * 2 * 40; i += bd) { // A biases -> whh row 10
    int s = i / 80, l = (i % 80) / 40, n = i % 40;
    lds[s * W_STK + (l ? 2304 : 1152) + fragIdx(10, permN(n))] = ba_ih[i] + ba_hh[i];
  }
  for (int i = tid; i < 3 * 40 * 20; i += bd) { // wb0_ih (3,40,20): remap K for padded concat
    int j = i / 800, r = i % 800, n = r / 20, k = r % 20;
    int kp = (k < 10) ? k : k + 2; // concat slices live at K 0..9 and 12..21
    lds[(4 + j) * W_STK + fragIdx(kp, permN(n))] = wb0_ih[i];
  }
  for (int i = tid; i < 3 * 40 * 10; i += bd) { // wb1_ih (3,40,10)
    int j = i / 400, r = i % 400, n = r / 10, k = r % 10;
    lds[(4 + j) * W_STK + 1728 + fragIdx(k, permN(n))] = wb1_ih[i];
  }
  for (int i = tid; i < 3 * 2 * 40 * 10; i += bd) { // wb_hh (3,2,40,10)
    int j = i / 800, r = i % 800, l = r / 400, n = (r % 400) / 10, k = r % 10;
    lds[(4 + j) * W_STK + (l ? 2304 : 1152) + fragIdx(k, permN(n))] = wb_hh[i];
  }
  for (int i = tid; i < 3 * 2 * 40; i += bd) { // B biases -> whh row 10
    int j = i / 80, l = (i % 80) / 40, n = i % 40;
    lds[(4 + j) * W_STK + (l ? 2304 : 1152) + fragIdx(10, permN(n))] = bb_ih[i] + bb_hh[i];
  }
  for (int i = tid; i < 100; i += bd) lds[20160 + i] = fc1_w[i];
  if (tid < 10) lds[20260 + tid] = fc1_b[tid];
  for (int i = tid; i < 60; i += bd) lds[20270 + i] = fc2_w[i];
  if (tid < 6) lds[20330 + tid] = fc2_b[tid];
  __syncthreads();

  const int lane = tid & 31;
  const int wave = tid >> 5;
  const int row = lane & 15;
  const int hi = lane >> 4;
  const int batch0 = blockIdx.x * 64 + wave * 16;

  float* P = lds + W_TOT + wave * PW_SZ;
  float* gA = P;
  float* gB = P + 832;
  float* hA0 = P + 1664; // + s*192, stride 12
  float* conU = P + 2432;
  float* conL = P + 2816;
  float* hB0 = P + 3200; // + u*192
  float* conB = P + 3584;
  float* hBo0 = P + 3968;
  float* hBo1 = P + 4160;

  // plant the constant-1.0 bias column in every h-state row (persists: activations
  // only write cols 0..9 / slice cols 0..9)
  if (lane < 16) {
    const int r = lane;
#pragma unroll
    for (int s = 0; s < 4; ++s) hA0[s * 192 + r * 12 + 10] = 1.f;
    conU[r * 24 + 10] = 1.f; conU[r * 24 + 22] = 1.f;
    conL[r * 24 + 10] = 1.f; conL[r * 24 + 22] = 1.f;
    hB0[r * 12 + 10] = 1.f;  hB0[192 + r * 12 + 10] = 1.f;
    conB[r * 24 + 10] = 1.f; conB[r * 24 + 22] = 1.f;
    hBo0[r * 12 + 10] = 1.f; hBo1[r * 12 + 10] = 1.f;
  }

  float cA[4][2][5], cB[2][2][5], cBo[2][5];
#pragma unroll
  for (int s = 0; s < 4; ++s)
#pragma unroll
    for (int l = 0; l < 2; ++l)
#pragma unroll
      for (int j = 0; j < 5; ++j) cA[s][l][j] = 0.f;
#pragma unroll
  for (int u = 0; u < 2; ++u)
#pragma unroll
    for (int l = 0; l < 2; ++l)
#pragma unroll
      for (int j = 0; j < 5; ++j) cB[u][l][j] = 0.f;
#pragma unroll
  for (int l = 0; l < 2; ++l)
#pragma unroll
    for (int j = 0; j < 5; ++j) cBo[l][j] = 0.f;

  const float* xbase = x + (size_t)(batch0 + row) * 5632; // T*D = 64*88

#pragma unroll 1
  for (int t = 0; t < 64; ++t) {
    const float* xt = xbase + t * 88;

    // ---- stage A layer0 (4 independent cells, alternating gate buffers) ----
#pragma unroll
    for (int s = 0; s < 4; ++s) {
      const float* W = lds + s * W_STK;
      const float* xrow = xt + 22 * s;
      float* h0 = hA0 + s * 192;
      lstm_cell<6>(
          [=](int kg, v2f& a) {
            int kk = 4 * kg + 2 * hi;
            if (kk > 20) kk = 20; // pad cols 22,23 have zero weights -> contribution 0
            a = *(const v2f*)(xrow + kk);
          },
          W, W + 1152, h0, 12, 0, (s & 1) ? gB : gA, h0, 12, 0,
          cA[s][0], row, hi, lane);
    }
    // ---- stage A layer1 -> concat slices (stride 24, slices at 0 / 12) ----
#pragma unroll
    for (int s = 0; s < 4; ++s) {
      const float* W = lds + s * W_STK;
      const float* h0 = hA0 + s * 192;
      float* cc = (s < 2) ? conU : conL;
      const int co = (s & 1) * 12;
      lstm_cell<3>(
          [=](int kg, v2f& a) { a = *(const v2f*)(h0 + row * 12 + 4 * kg + 2 * hi); },
          W + 1728, W + 2304, cc, 24, co, (s & 1) ? gB : gA, cc, 24, co,
          cA[s][1], row, hi, lane);
    }
    // ---- stage B layer0 (input = padded concat, K=24) ----
#pragma unroll
    for (int u = 0; u < 2; ++u) {
      const float* W = lds + (4 + u) * W_STK;
      const float* ci = u ? conL : conU;
      float* h0 = hB0 + u * 192;
      lstm_cell<6>(
          [=](int kg, v2f& a) { a = *(const v2f*)(ci + row * 24 + 4 * kg + 2 * hi); },
          W, W + 1152, h0, 12, 0, u ? gB : gA, h0, 12, 0,
          cB[u][0], row, hi, lane);
    }
    // ---- stage B layer1 -> concatBody slices ----
#pragma unroll
    for (int u = 0; u < 2; ++u) {
      const float* W = lds + (4 + u) * W_STK;
      const float* h0 = hB0 + u * 192;
      lstm_cell<3>(
          [=](int kg, v2f& a) { a = *(const v2f*)(h0 + row * 12 + 4 * kg + 2 * hi); },
          W + 1728, W + 2304, conB, 24, u * 12, u ? gB : gA, conB, 24, u * 12,
          cB[u][1], row, hi, lane);
    }
    // ---- body ----
    {
      const float* W = lds + 6 * W_STK;
      lstm_cell<6>(
          [=](int kg, v2f& a) { a = *(const v2f*)(conB + row * 24 + 4 * kg + 2 * hi); },
          W, W + 1152, hBo0, 12, 0, gA, hBo0, 12, 0, cBo[0], row, hi, lane);
      lstm_cell<3>(
          [=](int kg, v2f& a) { a = *(const v2f*)(hBo0 + row * 12 + 4 * kg + 2 * hi); },
          W + 1728, W + 2304, hBo1, 12, 0, gB, hBo1, 12, 0, cBo[1], row, hi, lane);
    }
  }

  // ---- FC head on last body h ----
  if (lane < 16) {
    const float* hb = hBo1 + lane * 12;
    const float* f1w = lds + 20160;
    const float* f1b = lds + 20260;
    const float* f2w = lds + 20270;
    const float* f2b = lds + 20330;
    float hv[10];
#pragma unroll
    for (int j = 0; j < 10; ++j) hv[j] = hb[j];
    float y[10];
#pragma unroll
    for (int o = 0; o < 10; ++o) {
      float a = f1b[o];
#pragma unroll
      for (int j = 0; j < 10; ++j) a += hv[j] * f1w[o * 10 + j];
      y[o] = a * sigm(a); // silu
    }
    float* op = out + (size_t)(batch0 + lane) * 6;
#pragma unroll
    for (int a2 = 0; a2 < 6; ++a2) {
      float acc = f2b[a2];
#pragma unroll
      for (int j = 0; j < 10; ++j) acc += y[j] * f2w[a2 * 10 + j];
      op[a2] = acc;
    }
  }
}

extern "C" void kernel_launch(void* const* d_in, const int* in_sizes, int n_in,
                              void* d_out, int out_size, void* d_ws, size_t ws_size,
                              hipStream_t stream) {
  (void)in_sizes; (void)n_in; (void)d_ws; (void)ws_size; (void)out_size;
  const float* x      = (const float*)d_in[0];
  const float* wa0_ih = (const float*)d_in[1];
  const float* wa1_ih = (const float*)d_in[2];
  const float* wa_hh  = (const float*)d_in[3];
  const float* ba_ih  = (const float*)d_in[4];
  const float* ba_hh  = (const float*)d_in[5];
  const float* wb0_ih = (const float*)d_in[6];
  const float* wb1_ih = (const float*)d_in[7];
  const float* wb_hh  = (const float*)d_in[8];
  const float* bb_ih  = (const float*)d_in[9];
  const float* bb_hh  = (const float*)d_in[10];
  const float* fc1_w  = (const float*)d_in[11];
  const float* fc1_b  = (const float*)d_in[12];
  const float* fc2_w  = (const float*)d_in[13];
  const float* fc2_b  = (const float*)d_in[14];
  float* out = (float*)d_out;

  (void)hipFuncSetAttribute((const void*)lstm_tree_kernel,
                            hipFuncAttributeMaxDynamicSharedMemorySize, LDS_BYTES);

  dim3 grid(256);  // 16384 rows / (4 waves * 16 rows)
  dim3 block(128); // 4 wave32 tiles per block
  lstm_tree_kernel<<<grid, block, LDS_BYTES, stream>>>(
      x, wa0_ih, wa1_ih, wa_hh, ba_ih, ba_hh,
      wb0_ih, wb1_ih, wb_hh, bb_ih, bb_hh,
      fc1_w, fc1_b, fc2_w, fc2_b, out);
}